// GATv2Encoder_49916109914173
// MI455X (gfx1250) — compile-verified
//
#include <hip/hip_runtime.h>
#include <hip/hip_bf16.h>
#include <math.h>

#define N_LAYERS 3
#define CDIM     128
#define EDIM     64
#define NSRC     200000
#define NDST     100000
#define NEDGE    600000
#define NEGSL    0.2f

typedef __attribute__((ext_vector_type(2))) float v2f;
typedef __attribute__((ext_vector_type(8))) float v8f;
typedef __attribute__((__vector_size__(4 * sizeof(int)))) int v4i;

__device__ __forceinline__ void atomicMaxF(float* addr, float v) {
  // classic monotonic-bits trick; buffer initialized to -inf
  if (v >= 0.0f) atomicMax((int*)addr, __float_as_int(v));
  else           atomicMin((unsigned int*)addr, __float_as_uint(v));
}

// ---------------------------------------------------------------------------
// Async global -> LDS stage (gfx1250 GLOBAL_LOAD_ASYNC_TO_LDS_B128, ASYNCcnt).
// Each active lane copies 16B from global address gsrc to LDS address sdst. Bypasses VGPRs.
// Builtin parameter types (from diagnostics): (int4 __device__*,
// int4 __shared__*, imm i32 offset, imm i32 cpol).
// ---------------------------------------------------------------------------
__device__ __forceinline__ void async_copy_b128(float* sdst, const float* gsrc) {
#if __has_builtin(__builtin_amdgcn_global_load_async_to_lds_b128)
  __attribute__((address_space(1))) v4i* g =
      (__attribute__((address_space(1))) v4i*)(unsigned long long)(uintptr_t)gsrc;
  __attribute__((address_space(3))) v4i* l =
      (__attribute__((address_space(3))) v4i*)(unsigned)(uintptr_t)sdst;
  __builtin_amdgcn_global_load_async_to_lds_b128(g, l, 0, 0);
#else
  asm volatile("global_load_async_to_lds_b128 %0, %1, off"
               :: "v"((unsigned)(uintptr_t)sdst),
                  "v"((unsigned long long)(uintptr_t)gsrc)
               : "memory");
#endif
}

__device__ __forceinline__ void wait_asynccnt0() {
#if __has_builtin(__builtin_amdgcn_s_wait_asynccnt)
  __builtin_amdgcn_s_wait_asynccnt(0);
#else
  asm volatile("s_wait_asynccnt 0" ::: "memory");
#endif
}

// Stage `nFloats` (multiple of 1024) floats from g into LDS sBuf, all 256
// threads participating; leaves ASYNCcnt drained for this wave.
__device__ __forceinline__ void stage_weights(float* sBuf, const float* g,
                                              int nFloats, int tid) {
  for (int i = tid * 4; i < nFloats; i += 256 * 4)
    async_copy_b128(sBuf + i, g + i);
  wait_asynccnt0();
}

// ---------------------------------------------------------------------------
// y[M x 128] = x[M x 128] @ W[128 x 128] + b   (fp32 WMMA 16x16x4)
// block = 256 threads (8 waves); W staged in LDS (64 KB); wave owns 16 rows.
// ---------------------------------------------------------------------------
__global__ __launch_bounds__(256) void gemm_node_kernel(
    const float* __restrict__ X, const float* __restrict__ Wg,
    const float* __restrict__ bias, float* __restrict__ Y, int M)
{
  __shared__ float sW[CDIM * CDIM];
  const int tid = threadIdx.x;
  stage_weights(sW, Wg, CDIM * CDIM, tid);
  __syncthreads();

  const int wave = tid >> 5;
  const int lane = tid & 31;
  const int l16  = lane & 15;
  const int hi   = lane >> 4;            // 0: K lanes {0,1}; 1: {2,3}
  const int row0 = blockIdx.x * 128 + wave * 16;
  if (row0 >= M) return;                 // wave-uniform: EXEC stays full

  int arow = row0 + l16;
  if (arow >= M) arow = M - 1;           // clamp tail rows (stores guarded)
  const float* xrow = X + (size_t)arow * CDIM;

  v8f acc[8] = {};
  #pragma unroll 2
  for (int k = 0; k < CDIM; k += 4) {
    float2 af = *(const float2*)(xrow + k + hi * 2);
    v2f a; a.x = af.x; a.y = af.y;
    #pragma unroll
    for (int t = 0; t < 8; ++t) {
      const int n = t * 16 + l16;
      v2f bf;
      bf.x = sW[(k + hi * 2) * CDIM + n];
      bf.y = sW[(k + hi * 2 + 1) * CDIM + n];
      acc[t] = __builtin_amdgcn_wmma_f32_16x16x4_f32(
                   false, a, false, bf, (short)0, acc[t], false, false);
    }
  }

  #pragma unroll
  for (int t = 0; t < 8; ++t) {
    const int n = t * 16 + l16;
    const float bn = bias[n];
    #pragma unroll
    for (int j = 0; j < 8; ++j) {
      const int m = row0 + j + hi * 8;   // C/D layout: VGPR j -> rows j, j+8
      if (m < M) Y[(size_t)m * CDIM + n] = acc[t][j] + bn;
    }
  }
}

// ---------------------------------------------------------------------------
// Fused edge pass 1: ew = edge_attr @ We (WMMA, We in LDS), then
// e = leaky(ew + xl[src] + xr[dst]); logits = e . att; atomicMax(mx[dst]).
// Never materializes the 600000 x 128 edge tensor.
// ---------------------------------------------------------------------------
__global__ __launch_bounds__(256) void edge_pass1_kernel(
    const float* __restrict__ EA, const float* __restrict__ Weg,
    const float* __restrict__ xl, const float* __restrict__ xr,
    const int* __restrict__ sidx, const int* __restrict__ didx,
    const float* __restrict__ att,
    float* __restrict__ logits, float* __restrict__ mx, int nE)
{
  __shared__ float sWe[EDIM * CDIM];
  const int tid = threadIdx.x;
  stage_weights(sWe, Weg, EDIM * CDIM, tid);
  __syncthreads();

  const int wave = tid >> 5;
  const int lane = tid & 31;
  const int l16  = lane & 15;
  const int hi   = lane >> 4;
  const int e0   = blockIdx.x * 128 + wave * 16;
  if (e0 >= nE) return;                  // wave-uniform

  int aerow = e0 + l16;
  if (aerow >= nE) aerow = nE - 1;
  const float* earow = EA + (size_t)aerow * EDIM;

  v8f acc[8] = {};
  #pragma unroll 2
  for (int k = 0; k < EDIM; k += 4) {
    float2 af = *(const float2*)(earow + k + hi * 2);
    v2f a; a.x = af.x; a.y = af.y;
    #pragma unroll
    for (int t = 0; t < 8; ++t) {
      const int n = t * 16 + l16;
      v2f bf;
      bf.x = sWe[(k + hi * 2) * CDIM + n];
      bf.y = sWe[(k + hi * 2 + 1) * CDIM + n];
      acc[t] = __builtin_amdgcn_wmma_f32_16x16x4_f32(
                   false, a, false, bf, (short)0, acc[t], false, false);
    }
  }

  float attv[8];
  #pragma unroll
  for (int t = 0; t < 8; ++t) attv[t] = att[t * 16 + l16];

  #pragma unroll
  for (int j = 0; j < 8; ++j) {
    const int m  = e0 + j + hi * 8;          // edge handled by this (j, half)
    const int me = (m < nE) ? m : (nE - 1);
    const int s = sidx[me];
    const int d = didx[me];
    const float* xls = xl + (size_t)s * CDIM;
    const float* xrd = xr + (size_t)d * CDIM;
    float partial = 0.0f;
    #pragma unroll
    for (int t = 0; t < 8; ++t) {
      const int n = t * 16 + l16;
      float ev = acc[t][j] + xls[n] + xrd[n];
      ev = (ev >= 0.0f) ? ev : NEGSL * ev;
      partial += ev * attv[t];
    }
    // reduce over the 16 lanes holding this row (xor masks stay in-half)
    partial += __shfl_xor(partial, 1, 32);
    partial += __shfl_xor(partial, 2, 32);
    partial += __shfl_xor(partial, 4, 32);
    partial += __shfl_xor(partial, 8, 32);
    if (l16 == 0 && m < nE) {
      logits[m] = partial;
      atomicMaxF(&mx[d], partial);
    }
  }
}

// ---------------------------------------------------------------------------
// Pass 2: ex = exp(logit - mx[dst]); den[dst] += ex  (ex stored in place)
// ---------------------------------------------------------------------------
__global__ void edge_pass2_kernel(float* __restrict__ logits,
                                  const float* __restrict__ mx,
                                  const int* __restrict__ didx,
                                  float* __restrict__ den, int nE)
{
  const int e = blockIdx.x * 256 + threadIdx.x;
  if (e >= nE) return;
  const int d = didx[e];
  const float ex = __expf(logits[e] - mx[d]);
  logits[e] = ex;
  atomicAdd(&den[d], ex);
}

// ---------------------------------------------------------------------------
// Pass 3: agg[dst] += (ex/(den[dst]+eps)) * xl[src]; one wave per edge,
// each lane owns 4 channels (float4 gather, 4 global float atomics).
// ---------------------------------------------------------------------------
__global__ __launch_bounds__(256) void edge_pass3_kernel(
    const float* __restrict__ ex, const float* __restrict__ den,
    const int* __restrict__ sidx, const int* __restrict__ didx,
    const float* __restrict__ xl, float* __restrict__ agg, int nE)
{
  const int lane  = threadIdx.x & 31;
  const int wave  = (blockIdx.x * 256 + threadIdx.x) >> 5;
  const int nWave = gridDim.x * 8;
  for (int e = wave; e < nE; e += nWave) {
    const int s = sidx[e];
    const int d = didx[e];
    const float alpha = ex[e] / (den[d] + 1e-16f);
    const float4 v = ((const float4*)(xl + (size_t)s * CDIM))[lane];
    float* ad = agg + (size_t)d * CDIM + lane * 4;
    atomicAdd(ad + 0, alpha * v.x);
    atomicAdd(ad + 1, alpha * v.y);
    atomicAdd(ad + 2, alpha * v.z);
    atomicAdd(ad + 3, alpha * v.w);
  }
}

__global__ void init_layer_kernel(float* __restrict__ agg, float* __restrict__ mx,
                                  float* __restrict__ den, int nDst)
{
  const int i = blockIdx.x * 256 + threadIdx.x;
  if (i < nDst * CDIM) agg[i] = 0.0f;
  if (i < nDst) { mx[i] = -INFINITY; den[i] = 0.0f; }
}

__global__ void finalize_kernel(const float* __restrict__ agg,
                                const float* __restrict__ bias_out,
                                const float* __restrict__ pa,
                                float* __restrict__ out, int nDst)
{
  const int i = blockIdx.x * 256 + threadIdx.x;
  if (i >= nDst * CDIM) return;
  const int c = i & (CDIM - 1);
  const float v = agg[i] + bias_out[c];
  out[i] = (v >= 0.0f) ? v : pa[c] * v;
}

// ---------------------------------------------------------------------------
extern "C" void kernel_launch(void* const* d_in, const int* in_sizes, int n_in,
                              void* d_out, int out_size, void* d_ws, size_t ws_size,
                              hipStream_t stream)
{
  const float* x_src    = (const float*)d_in[0];
  const float* x_dst    = (const float*)d_in[1];
  const int*   src_idx  = (const int*)d_in[2];
  const int*   dst_idx  = (const int*)d_in[3];
  const float* edge_att = (const float*)d_in[4];
  const float* W        = (const float*)d_in[5];
  const float* b        = (const float*)d_in[6];
  const float* We       = (const float*)d_in[7];
  const float* att      = (const float*)d_in[8];
  const float* bias_out = (const float*)d_in[9];
  const float* prelu_a  = (const float*)d_in[10];
  (void)in_sizes; (void)n_in; (void)out_size; (void)ws_size;

  float* out = (float*)d_out;

  // workspace partition
  float* ws     = (float*)d_ws;
  float* xl     = ws;                                  // NSRC*128
  float* xr     = xl     + (size_t)NSRC * CDIM;        // NDST*128
  float* agg    = xr     + (size_t)NDST * CDIM;        // NDST*128
  float* logits = agg    + (size_t)NDST * CDIM;        // NEDGE
  float* mx     = logits + NEDGE;                      // NDST
  float* den    = mx     + NDST;                       // NDST

  const int gGemmSrc = (NSRC + 127) / 128;
  const int gGemmDst = (NDST + 127) / 128;
  const int gEdge1   = (NEDGE + 127) / 128;
  const int gEdge2   = (NEDGE + 255) / 256;
  const int gElem    = (NDST * CDIM + 255) / 256;

  const float* x_r = x_dst;
  for (int layer = 0; layer < N_LAYERS; ++layer) {
    const float* Wi  = W        + (size_t)layer * CDIM * CDIM;
    const float* bi  = b        + (size_t)layer * CDIM;
    const float* Wei = We       + (size_t)layer * EDIM * CDIM;
    const float* ai  = att      + (size_t)layer * CDIM;
    const float* boi = bias_out + (size_t)layer * CDIM;

    init_layer_kernel<<<gElem, 256, 0, stream>>>(agg, mx, den, NDST);
    gemm_node_kernel<<<gGemmSrc, 256, 0, stream>>>(x_src, Wi, bi, xl, NSRC);
    gemm_node_kernel<<<gGemmDst, 256, 0, stream>>>(x_r,   Wi, bi, xr, NDST);
    edge_pass1_kernel<<<gEdge1, 256, 0, stream>>>(edge_att, Wei, xl, xr,
                                                  src_idx, dst_idx, ai,
                                                  logits, mx, NEDGE);
    edge_pass2_kernel<<<gEdge2, 256, 0, stream>>>(logits, mx, dst_idx, den, NEDGE);
    edge_pass3_kernel<<<gEdge2, 256, 0, stream>>>(logits, den, src_idx, dst_idx,
                                                  xl, agg, NEDGE);
    finalize_kernel<<<gElem, 256, 0, stream>>>(agg, boi, prelu_a, out, NDST);

    x_r = out;   // next layer reads this layer's output
  }
}